// Density_loss_37065567765211
// MI455X (gfx1250) — compile-verified
//
#include <hip/hip_runtime.h>
#include <hip/hip_bf16.h>

// ---------------------------------------------------------------------------
// Density loss: fused fp32 GEMM (WMMA f32 16x16x4) + sqrt-distance + weights
// + per-row top-5 smallest + hinge mean.  N=M=8192, D=512, k=5.
// Round 3: fix LDS pointer-array static-initializer issue (compute buffer
// base from byte offset per use instead of an array of LDS pointers).
// ---------------------------------------------------------------------------

typedef float v2f __attribute__((ext_vector_type(2)));
typedef float v8f __attribute__((ext_vector_type(8)));

#define NN    8192
#define MM    8192
#define DD    512
#define ROWS  32           // rows per workgroup (strip)
#define COLS  128          // columns per macro tile (8 waves x 16r x 32c)
#define KCH   64           // K-chunk per async-staged B buffer
#define NCH   (DD / KCH)   // 8 chunks per tile
#define TOPK  5
#define ASTR  (DD + 4)     // A LDS stride (516 fl: 258 ≡ 2 mod 32 -> cf b64)
#define BSTR  (KCH/2 + 2)  // B LDS stride in float2 (34 ≡ 2 mod 32 -> cf b64)
#define WSTR  (COLS + 1)   // wd tile stride (129 ≡ 1 mod 64 -> cf row scan)

// LDS byte offsets within dynamic shared
#define OFF_AS   0u
#define OFF_BS0  (OFF_AS  + (unsigned)(ROWS * ASTR * 4))        // 66048
#define OFF_BS1  (OFF_BS0 + (unsigned)(COLS * BSTR * 8))        // +34816
#define OFF_WD   (OFF_BS1 + (unsigned)(COLS * BSTR * 8))        // +34816
#define OFF_ROWF (OFF_WD  + (unsigned)(ROWS * WSTR * 4))        // +16512
#define OFF_RSN  (OFF_ROWF + (unsigned)(ROWS * 4))
#define SMEM_BYTES (OFF_RSN + (unsigned)(ROWS * 4))

// ---------------- single-block deterministic vector sum --------------------
__global__ void vec_sum_kernel(const float* __restrict__ x, int n,
                               float* __restrict__ out) {
  __shared__ float red[256];
  float s = 0.f;
  for (int i = threadIdx.x; i < n; i += 256) s += x[i];
  red[threadIdx.x] = s;
  __syncthreads();
  for (int st = 128; st; st >>= 1) {
    if ((int)threadIdx.x < st) red[threadIdx.x] += red[threadIdx.x + st];
    __syncthreads();
  }
  if (threadIdx.x == 0) out[0] = red[0];
}

// ---------------- per-row squared norms (one wave32 per row) ---------------
__global__ void row_norms_kernel(const float* __restrict__ x,
                                 float* __restrict__ out, int rows) {
  int row  = blockIdx.x * 8 + (threadIdx.x >> 5);
  int lane = threadIdx.x & 31;
  if (row >= rows) return;
  const float* p = x + (size_t)row * DD;
  float s = 0.f;
#pragma unroll
  for (int i = 0; i < 4; ++i) {
    float4 v = *(const float4*)&p[lane * 4 + i * 128];
    s += v.x * v.x + v.y * v.y + v.z * v.z + v.w * v.w;
  }
#pragma unroll
  for (int off = 16; off; off >>= 1) s += __shfl_xor(s, off, 32);
  if (lane == 0) out[row] = s;
}

// ---------------- main fused kernel ----------------------------------------
__global__ __launch_bounds__(256) void density_main_kernel(
    const float* __restrict__ src,   // [N, D]
    const float* __restrict__ tgt,   // [M, D]
    const float* __restrict__ swraw, // [N]
    const float* __restrict__ twraw, // [M]
    const float* __restrict__ sums,  // [2] : sum(sw), sum(tw)
    const float* __restrict__ sn,    // [N] row norms
    const float* __restrict__ tn,    // [M] row norms
    float* __restrict__ partials)    // [gridDim.x] hinge partial sums
{
  extern __shared__ char smem_raw[];
  float* As   = (float*)(smem_raw + OFF_AS);
  float* Wd   = (float*)(smem_raw + OFF_WD);
  float* rowf = (float*)(smem_raw + OFF_ROWF);
  float* rsn  = (float*)(smem_raw + OFF_RSN);

  const int tid  = threadIdx.x;
  const int wid  = tid >> 5;
  const int lane = tid & 31;
  const int ln   = lane & 15;
  const int h    = lane >> 4;      // lane half
  const int rh   = wid >> 2;       // row half (0..1)
  const int cp   = wid & 3;        // col pair (0..3), covers 32 columns
  const int rowbase = blockIdx.x * ROWS;

  const float scaleS = (float)NN / (sums[0] + 1e-8f);
  const float scaleT = (float)MM / (sums[1] + 1e-8f);

  // ---- stage A strip: 32 rows x 512 K into LDS (coalesced float4) ----
  for (int i = tid; i < ROWS * (DD / 4); i += 256) {
    int r  = i / (DD / 4);
    int k4 = (i % (DD / 4)) * 4;
    float4 v = *(const float4*)&src[(size_t)(rowbase + r) * DD + k4];
    *(float4*)&As[r * ASTR + k4] = v;
  }
  if (tid < ROWS) {
    rowf[tid] = swraw[rowbase + tid] * scaleS;
    rsn[tid]  = sn[rowbase + tid];
  }
  __syncthreads();

  // per-row top-5 (ascending, t4 = current 5th smallest), lives in wave 0
  const float FINF = 3.4e38f;
  float t0 = FINF, t1 = FINF, t2 = FINF, t3 = FINF, t4 = FINF;

  for (int cbase = 0; cbase < MM; cbase += COLS) {
    v8f acc0 = {0.f, 0.f, 0.f, 0.f, 0.f, 0.f, 0.f, 0.f};
    v8f acc1 = {0.f, 0.f, 0.f, 0.f, 0.f, 0.f, 0.f, 0.f};

    // async-stage helper: one KCH chunk of B (COLS x KCH floats) into buffer
    // 2048 x 16B transfers, 8 per thread, fully coalesced along K.
    auto stageB = [&](unsigned bufOff, int kk) {
#pragma unroll
      for (int j = 0; j < (COLS * (KCH / 4)) / 256; ++j) {  // 8 iters
        int linear = j * 256 + tid;
        int c = linear >> 4;        // 0..127
        int q = linear & 15;        // 0..15 -> k = 4q
        unsigned ldsa = bufOff + (unsigned)(c * BSTR + 2 * q) * 8u;
        const float* gp = tgt + (size_t)(cbase + c) * DD + (kk + 4 * q);
        asm volatile("global_load_async_to_lds_b128 %0, %1, off"
                     :: "v"(ldsa), "v"(gp) : "memory");
      }
    };

    stageB(OFF_BS0, 0);  // prologue: chunk 0 into buffer 0

    for (int kc = 0; kc < NCH; ++kc) {
      // drain my async issues for chunk kc, then make all waves' LDS visible
      asm volatile("s_wait_asynccnt 0x0" ::: "memory");
      __syncthreads();
      if (kc + 1 < NCH)
        stageB(((kc + 1) & 1) ? OFF_BS1 : OFF_BS0, (kc + 1) * KCH);

      // ---- WMMA f32 16x16x4 over chunk kc (shared A, two B tiles) ----
      const float2* Bsbuf =
          (const float2*)(smem_raw + ((kc & 1) ? OFF_BS1 : OFF_BS0));
      const float*  arow = &As[(rh * 16 + ln) * ASTR + kc * KCH];
      const float2* b0   = &Bsbuf[(cp * 32 + ln) * BSTR];
      const float2* b1   = b0 + 16 * BSTR;
#pragma unroll
      for (int k4 = 0; k4 < KCH; k4 += 4) {
        v2f a   = *(const v2f*)&arow[k4 + 2 * h];       // A: K = k4+2h+v
        v2f bb0 = *(const v2f*)&b0[(k4 >> 1) + h];      // B: K = k4+2h+v
        v2f bb1 = *(const v2f*)&b1[(k4 >> 1) + h];
        acc0 = __builtin_amdgcn_wmma_f32_16x16x4_f32(
            false, a, false, bb0, (short)0, acc0, false, false);
        acc1 = __builtin_amdgcn_wmma_f32_16x16x4_f32(
            false, a, false, bb1, (short)0, acc1, false, false);
      }
    }

    // ---- epilogue: weighted sqrt distance into LDS tile ----
    const int gc0 = cbase + cp * 32 + ln;
    const int gc1 = gc0 + 16;
    const float tnv0  = tn[gc0];
    const float tnv1  = tn[gc1];
    const float colf0 = twraw[gc0] * scaleT;
    const float colf1 = twraw[gc1] * scaleT;
#pragma unroll
    for (int v = 0; v < 8; ++v) {
      int lr = rh * 16 + v + 8 * h;  // C layout: M = v + 8*(lane>=16)
      float fr = rowf[lr], nr = rsn[lr];
      float d20 = nr + tnv0 - 2.0f * acc0[v];
      float d21 = nr + tnv1 - 2.0f * acc1[v];
      float wd0 = __builtin_sqrtf(fmaxf(d20, 0.0f)) * fr * colf0;
      float wd1 = __builtin_sqrtf(fmaxf(d21, 0.0f)) * fr * colf1;
      Wd[lr * WSTR + cp * 32 + ln]      = wd0;
      Wd[lr * WSTR + cp * 32 + 16 + ln] = wd1;
    }
    __syncthreads();

    // ---- wave 0: one lane per row scans 128 candidates, keeps 5 smallest --
    if (tid < ROWS) {
      const float* wr = &Wd[tid * WSTR];
#pragma unroll 4
      for (int j = 0; j < COLS; ++j) {
        float v = wr[j];
        if (v < t4) {
          t4 = v;
          if (t4 < t3) { float t = t3; t3 = t4; t4 = t; }
          if (t3 < t2) { float t = t2; t2 = t3; t3 = t; }
          if (t2 < t1) { float t = t1; t1 = t2; t2 = t; }
          if (t1 < t0) { float t = t0; t0 = t1; t1 = t; }
        }
      }
    }
    __syncthreads();  // Wd + Bs buffers free for next tile
  }

  // ---- hinge + deterministic per-block partial ----
  if (tid < ROWS) {
    const float hv = 0.01f;
    float s = fmaxf(t0 - hv, 0.f) + fmaxf(t1 - hv, 0.f) + fmaxf(t2 - hv, 0.f) +
              fmaxf(t3 - hv, 0.f) + fmaxf(t4 - hv, 0.f);
#pragma unroll
    for (int off = 16; off; off >>= 1) s += __shfl_xor(s, off, 32);
    if (lane == 0) partials[blockIdx.x] = s;
  }
}

// ---------------- deterministic final reduction ----------------------------
__global__ void finalize_kernel(const float* __restrict__ partials, int n,
                                float* __restrict__ out) {
  __shared__ float red[256];
  float s = ((int)threadIdx.x < n) ? partials[threadIdx.x] : 0.f;
  red[threadIdx.x] = s;
  __syncthreads();
  for (int st = 128; st; st >>= 1) {
    if ((int)threadIdx.x < st) red[threadIdx.x] += red[threadIdx.x + st];
    __syncthreads();
  }
  if (threadIdx.x == 0)
    out[0] = red[0] * (1.0f / ((float)NN * (float)TOPK));
}

// ---------------------------------------------------------------------------
extern "C" void kernel_launch(void* const* d_in, const int* in_sizes, int n_in,
                              void* d_out, int out_size, void* d_ws, size_t ws_size,
                              hipStream_t stream) {
  const float* src = (const float*)d_in[0];  // source   [N, D]
  const float* tgt = (const float*)d_in[1];  // target   [M, D]
  const float* sw  = (const float*)d_in[2];  // source_weights [N]
  const float* tw  = (const float*)d_in[3];  // target_weights [M]
  // d_in[4] = top_k (==5, compile-time constant here)
  (void)in_sizes; (void)n_in; (void)out_size;

  float* ws       = (float*)d_ws;
  float* sums     = ws;                 // [2]
  float* sn       = ws + 8;             // [N]
  float* tn       = sn + NN;            // [M]
  float* partials = tn + MM;            // [N/ROWS]
  (void)ws_size;

  // deterministic weight sums (single block each)
  vec_sum_kernel<<<1, 256, 0, stream>>>(sw, NN, &sums[0]);
  vec_sum_kernel<<<1, 256, 0, stream>>>(tw, MM, &sums[1]);

  // row norms (one wave per row)
  row_norms_kernel<<<NN / 8, 256, 0, stream>>>(src, sn, NN);
  row_norms_kernel<<<MM / 8, 256, 0, stream>>>(tgt, tn, MM);

  const int nblocks = NN / ROWS;  // 256
  density_main_kernel<<<nblocks, 256, (size_t)SMEM_BYTES, stream>>>(
      src, tgt, sw, tw, sums, sn, tn, partials);

  finalize_kernel<<<1, 256, 0, stream>>>(partials, nblocks, (float*)d_out);
}